// PWGNN_73839077753372
// MI455X (gfx1250) — compile-verified
//
#include <hip/hip_runtime.h>
#include <math.h>

typedef float v2f __attribute__((ext_vector_type(2)));
typedef float v4f __attribute__((ext_vector_type(4)));
typedef float v8f __attribute__((ext_vector_type(8)));

// ---------------------------------------------------------------------------
// Kernel 1: per-node message via WMMA (fp32, no precision loss).
//   x = unary/deg ; M_r = max_c x[r][c] ; Ex = exp(x - M_r)
//   P = Ex @ expB       (16x8 @ 8x16-padded, two V_WMMA_F32_16X16X4_F32)
//   msg[r][c2] = M_r + log(P[r][c2])
// One wave32 = 16 nodes. EXEC stays full through the WMMA path (only
// wave-uniform early-out); divergence only at the final stores.
// Also zero-initializes aggr for its 16 nodes (workspace is poisoned).
// ---------------------------------------------------------------------------
__global__ __launch_bounds__(256) void pwgnn_node_msg(
    const float* __restrict__ unary, const float* __restrict__ deg,
    const float* __restrict__ binary, float* __restrict__ msg,
    float* __restrict__ aggr, int N)
{
    __shared__ float sB[64];                      // exp(B_half), 8x8
    int tid = threadIdx.x;
    if (tid < 64) {
        int i = tid >> 3, j = tid & 7;
        // B_half = (binary + binary^T) * 0.25
        sB[tid] = __expf((binary[i * 8 + j] + binary[j * 8 + i]) * 0.25f);
    }
    __syncthreads();

    int lane = tid & 31;
    int wave = tid >> 5;
    int nb = (blockIdx.x * (blockDim.x >> 5) + wave) * 16;   // first node of tile
    if (nb >= N) return;                                     // wave-uniform

    int r   = lane & 15;        // node row within tile (also output column)
    bool hi = (lane & 16) != 0; // half-wave selector (K split per ISA layout)

    int node = min(nb + r, N - 1);
    float dinv = 1.0f / deg[node];
    const v4f* up = (const v4f*)(unary + (size_t)node * 8);
    v4f x0 = up[0], x1 = up[1];

    float e0 = x0.x * dinv, e1 = x0.y * dinv, e2 = x0.z * dinv, e3 = x0.w * dinv;
    float e4 = x1.x * dinv, e5 = x1.y * dinv, e6 = x1.z * dinv, e7 = x1.w * dinv;
    float M = fmaxf(fmaxf(fmaxf(e0, e1), fmaxf(e2, e3)),
                    fmaxf(fmaxf(e4, e5), fmaxf(e6, e7)));
    e0 = __expf(e0 - M); e1 = __expf(e1 - M); e2 = __expf(e2 - M); e3 = __expf(e3 - M);
    e4 = __expf(e4 - M); e5 = __expf(e5 - M); e6 = __expf(e6 - M); e7 = __expf(e7 - M);

    // A-matrix 16x4 f32 layout: lanes 0-15 -> {K0,K1}, lanes 16-31 -> {K2,K3}.
    // Single-bit selects (v_cndmask), no dynamic register indexing.
    v2f a0 = { hi ? e2 : e0, hi ? e3 : e1 };      // K-block 0..3
    v2f a1 = { hi ? e6 : e4, hi ? e7 : e5 };      // K-block 4..7

    // B-matrix 4x16 f32, rows striped across lanes; columns 8..15 padded 0.
    int  col   = r;
    bool incol = col < 8;
    int  kb    = hi ? 2 : 0;
    v2f b0 = { incol ? sB[(kb + 0) * 8 + col] : 0.0f,
               incol ? sB[(kb + 1) * 8 + col] : 0.0f };
    v2f b1 = { incol ? sB[(kb + 4) * 8 + col] : 0.0f,
               incol ? sB[(kb + 5) * 8 + col] : 0.0f };

    v8f acc = {};
    acc = __builtin_amdgcn_wmma_f32_16x16x4_f32(false, a0, false, b0,
                                                (short)0, acc, false, false);
    acc = __builtin_amdgcn_wmma_f32_16x16x4_f32(false, a1, false, b1,
                                                (short)0, acc, false, false);

    // Broadcast per-row maxes BEFORE any divergent branch (ds_bpermute reads
    // zero from EXEC-disabled lanes). Lane q holds the max of row q (q<16).
    int rbase = hi ? 8 : 0;
    float Mrow[8];
    #pragma unroll
    for (int v = 0; v < 8; ++v) Mrow[v] = __shfl(M, rbase + v);

    if (nb + 16 <= N) {                            // fast path: full tile
        if (incol) {
            #pragma unroll
            for (int v = 0; v < 8; ++v) {
                int row = nb + rbase + v;          // D layout: M = v + 8*hi
                msg[(size_t)row * 8 + col] = Mrow[v] + __logf(acc[v]);
            }
        }
        #pragma unroll
        for (int q = 0; q < 4; ++q)
            aggr[(size_t)nb * 8 + q * 32 + lane] = 0.0f;
    } else {                                       // cold tail path
        if (incol) {
            #pragma unroll
            for (int v = 0; v < 8; ++v) {
                int row = nb + rbase + v;
                if (row < N)
                    msg[(size_t)row * 8 + col] = Mrow[v] + __logf(acc[v]);
            }
        }
        #pragma unroll
        for (int q = 0; q < 4; ++q) {
            size_t idx = (size_t)nb * 8 + q * 32 + lane;
            if (idx < (size_t)N * 8) aggr[idx] = 0.0f;
        }
    }
}

// ---------------------------------------------------------------------------
// Kernel 2: edge scatter. aggr[src] += msg[dst].  Both tables are L2-resident
// (3.2 MB each vs 192 MB L2); hardware fp32 atomics, no CAS loop.
// ---------------------------------------------------------------------------
__global__ __launch_bounds__(256) void pwgnn_edge_scatter(
    const int* __restrict__ ei, const float* __restrict__ msg,
    float* __restrict__ aggr, int E)
{
    int t = blockIdx.x * blockDim.x + threadIdx.x;
    if (t >= E) return;
    int s = ei[t];          // edge_index[0] : scatter target
    int d = ei[E + t];      // edge_index[1] : gather source
    const v4f* mp = (const v4f*)(msg + (size_t)d * 8);
    v4f a = mp[0], b = mp[1];
    float* ap = aggr + (size_t)s * 8;
    __hip_atomic_fetch_add(ap + 0, a.x, __ATOMIC_RELAXED, __HIP_MEMORY_SCOPE_AGENT);
    __hip_atomic_fetch_add(ap + 1, a.y, __ATOMIC_RELAXED, __HIP_MEMORY_SCOPE_AGENT);
    __hip_atomic_fetch_add(ap + 2, a.z, __ATOMIC_RELAXED, __HIP_MEMORY_SCOPE_AGENT);
    __hip_atomic_fetch_add(ap + 3, a.w, __ATOMIC_RELAXED, __HIP_MEMORY_SCOPE_AGENT);
    __hip_atomic_fetch_add(ap + 4, b.x, __ATOMIC_RELAXED, __HIP_MEMORY_SCOPE_AGENT);
    __hip_atomic_fetch_add(ap + 5, b.y, __ATOMIC_RELAXED, __HIP_MEMORY_SCOPE_AGENT);
    __hip_atomic_fetch_add(ap + 6, b.z, __ATOMIC_RELAXED, __HIP_MEMORY_SCOPE_AGENT);
    __hip_atomic_fetch_add(ap + 7, b.w, __ATOMIC_RELAXED, __HIP_MEMORY_SCOPE_AGENT);
}

// ---------------------------------------------------------------------------
// Kernel 3: per-node log partition: log_z = LSE_c(unary/deg + aggr)
// ---------------------------------------------------------------------------
__global__ __launch_bounds__(256) void pwgnn_node_lse(
    const float* __restrict__ unary, const float* __restrict__ deg,
    const float* __restrict__ aggr, float* __restrict__ out, int N)
{
    int n = blockIdx.x * blockDim.x + threadIdx.x;
    if (n >= N) return;
    float dinv = 1.0f / deg[n];
    const v4f* up = (const v4f*)(unary + (size_t)n * 8);
    const v4f* gp = (const v4f*)(aggr  + (size_t)n * 8);
    v4f u0 = up[0], u1 = up[1], g0 = gp[0], g1 = gp[1];
    float t0 = u0.x * dinv + g0.x, t1 = u0.y * dinv + g0.y;
    float t2 = u0.z * dinv + g0.z, t3 = u0.w * dinv + g0.w;
    float t4 = u1.x * dinv + g1.x, t5 = u1.y * dinv + g1.y;
    float t6 = u1.z * dinv + g1.z, t7 = u1.w * dinv + g1.w;
    float M = fmaxf(fmaxf(fmaxf(t0, t1), fmaxf(t2, t3)),
                    fmaxf(fmaxf(t4, t5), fmaxf(t6, t7)));
    float s = __expf(t0 - M) + __expf(t1 - M) + __expf(t2 - M) + __expf(t3 - M)
            + __expf(t4 - M) + __expf(t5 - M) + __expf(t6 - M) + __expf(t7 - M);
    out[n] = M + __logf(s);
}

extern "C" void kernel_launch(void* const* d_in, const int* in_sizes, int n_in,
                              void* d_out, int out_size, void* d_ws, size_t ws_size,
                              hipStream_t stream)
{
    const float* unary  = (const float*)d_in[0];   // [N,8]
    const float* deg    = (const float*)d_in[1];   // [N,1]
    const float* binary = (const float*)d_in[2];   // [8,8]
    const int*   ei     = (const int*)d_in[3];     // [2,E]
    float* out = (float*)d_out;                    // [N]

    int N = in_sizes[1];        // deg has N elements
    int E = in_sizes[3] / 2;

    float* msg  = (float*)d_ws;                    // [N,8]  3.2 MB
    float* aggr = msg + (size_t)N * 8;             // [N,8]  3.2 MB

    int nodesPerBlock = (256 / 32) * 16;           // 8 waves x 16 nodes = 128
    pwgnn_node_msg<<<(N + nodesPerBlock - 1) / nodesPerBlock, 256, 0, stream>>>(
        unary, deg, binary, msg, aggr, N);
    pwgnn_edge_scatter<<<(E + 255) / 256, 256, 0, stream>>>(ei, msg, aggr, E);
    pwgnn_node_lse<<<(N + 255) / 256, 256, 0, stream>>>(unary, deg, aggr, out, N);
}